// FPSampler_30897994728113
// MI455X (gfx1250) — compile-verified
//
#include <hip/hip_runtime.h>
#include <hip/hip_bf16.h>
#include <stdint.h>

#define N_POINTS  131072
#define N_SAMPLE  4096
#define NBLK      32
#define TPB       256
#define NTHREADS  (NBLK * TPB)            // 8192
#define PPT       (N_POINTS / NTHREADS)   // 16 points per thread, register-resident
#define NWAVES    (TPB / 32)              // 8 wave32s per block

// ---------------------------------------------------------------------------
// Scratch layout in d_ws:
//   [0..23]    : 3 x u64 round-robin winner slots (packed (d_bits<<32)|~idx)
//   [128]      : grid-barrier arrival counter
//   [256]      : grid-barrier generation counter
// ---------------------------------------------------------------------------

__device__ __forceinline__ void grid_barrier(unsigned* cnt, unsigned* gen, unsigned nb) {
    __syncthreads();
    if (threadIdx.x == 0) {
        unsigned g = __hip_atomic_load(gen, __ATOMIC_RELAXED, __HIP_MEMORY_SCOPE_AGENT);
        unsigned a = __hip_atomic_fetch_add(cnt, 1u, __ATOMIC_ACQ_REL, __HIP_MEMORY_SCOPE_AGENT);
        if (a == nb - 1u) {
            // all blocks arrived: reset counter for next use, then release generation
            __hip_atomic_store(cnt, 0u, __ATOMIC_RELAXED, __HIP_MEMORY_SCOPE_AGENT);
            __hip_atomic_fetch_add(gen, 1u, __ATOMIC_ACQ_REL, __HIP_MEMORY_SCOPE_AGENT);
        } else {
            while (__hip_atomic_load(gen, __ATOMIC_ACQUIRE, __HIP_MEMORY_SCOPE_AGENT) == g) {
                __builtin_amdgcn_s_sleep(1);
            }
        }
    }
    __syncthreads();
}

// Persistent FPS kernel: pos tile + running min-distance live in VGPRs for the
// whole 4096-step serial chain; each step costs one 12B async LDS broadcast,
// ~200 VALU ops/thread, one packed u64 atomicMax per block, one grid barrier.
__global__ __launch_bounds__(TPB, 1)
void fps_persistent_kernel(const float* __restrict__ pos, int* __restrict__ out,
                           unsigned long long* slots, unsigned* cnt, unsigned* gen) {
    const int tid = threadIdx.x;
    const int g   = blockIdx.x * TPB + tid;

    __shared__ float    qsh[4];       // broadcast query point (async-to-LDS target)
    __shared__ float    swv[NWAVES];  // per-wave best value
    __shared__ unsigned swi[NWAVES];  // per-wave best index

    // ---- one-time load: this thread's 16 points + running min-dist, all in VGPRs
    float px[PPT], py[PPT], pz[PPT], d[PPT];
#pragma unroll
    for (int p = 0; p < PPT; ++p) {
        const int idx = g + p * NTHREADS;
        px[p] = pos[3 * idx + 0];
        py[p] = pos[3 * idx + 1];
        pz[p] = pos[3 * idx + 2];
        d[p]  = __builtin_inff();
    }

    unsigned last = 0u;  // random_start=False -> start index 0

    for (int i = 0; i < N_SAMPLE; ++i) {
        if (g == 0) out[i] = (int)last;  // scan emits `last` BEFORE this step's pick

        // ---- broadcast pos[last] into LDS via CDNA5 async global->LDS copy.
        // Lanes 0..2 of wave 0 each move one 4B component; ASYNCcnt-tracked.
        if (tid < 3) {
            unsigned lds_off =
                (unsigned)(uintptr_t)(&qsh[tid]);  // generic LDS addr low 32b = offset
            unsigned long long gaddr =
                (unsigned long long)(uintptr_t)(pos + 3ull * (unsigned long long)last + (unsigned)tid);
            asm volatile(
                "global_load_async_to_lds_b32 %0, %1, off\n\t"
                "s_wait_asynccnt 0"
                :: "v"(lds_off), "v"(gaddr)
                : "memory");
        }
        __syncthreads();
        const float qx = qsh[0], qy = qsh[1], qz = qsh[2];

        // ---- distance + running min + thread-local argmax (strict > => lowest idx)
        // __f*_rn prevents FMA contraction: matches XLA's mul-then-reduce lowering.
        float    bv = -__builtin_inff();
        unsigned bi = 0u;
#pragma unroll
        for (int p = 0; p < PPT; ++p) {
            const float dx   = __fsub_rn(px[p], qx);
            const float dy   = __fsub_rn(py[p], qy);
            const float dz   = __fsub_rn(pz[p], qz);
            const float dist = __fadd_rn(__fadd_rn(__fmul_rn(dx, dx), __fmul_rn(dy, dy)),
                                         __fmul_rn(dz, dz));
            const float nd   = fminf(d[p], dist);
            d[p] = nd;
            if (nd > bv) { bv = nd; bi = (unsigned)(g + p * NTHREADS); }
        }

        // ---- wave32 argmax reduction (ties -> lowest index)
#pragma unroll
        for (int o = 16; o > 0; o >>= 1) {
            const float    ov = __shfl_xor(bv, o, 32);
            const unsigned oi = __shfl_xor(bi, o, 32);
            if (ov > bv || (ov == bv && oi < bi)) { bv = ov; bi = oi; }
        }
        if ((tid & 31) == 0) { swv[tid >> 5] = bv; swi[tid >> 5] = bi; }
        __syncthreads();

        // ---- block winner -> one packed u64 atomicMax per block
        if (tid == 0) {
            float    cbv = swv[0];
            unsigned cbi = swi[0];
#pragma unroll
            for (int w = 1; w < NWAVES; ++w) {
                const float    ov = swv[w];
                const unsigned oi = swi[w];
                if (ov > cbv || (ov == cbv && oi < cbi)) { cbv = ov; cbi = oi; }
            }
            const unsigned long long key =
                ((unsigned long long)__float_as_uint(cbv) << 32) |
                (unsigned long long)(0xFFFFFFFFu - cbi);
            atomicMax(&slots[i % 3], key);
        }
        __syncthreads();

        grid_barrier(cnt, gen, NBLK);

        // Retire the slot last read before the barrier we just crossed; its next
        // writer (iteration i+2) is separated from this store by barrier i+1.
        if (g == 0) {
            __hip_atomic_store(&slots[(i + 2) % 3], 0ull,
                               __ATOMIC_RELAXED, __HIP_MEMORY_SCOPE_AGENT);
        }
        const unsigned long long wkey =
            __hip_atomic_load(&slots[i % 3], __ATOMIC_RELAXED, __HIP_MEMORY_SCOPE_AGENT);
        last = 0xFFFFFFFFu - (unsigned)(wkey & 0xFFFFFFFFull);
    }
}

// Prologue kernel: deterministically clears the winner slots and barrier words
// each call (d_ws is poisoned once and never re-poisoned between graph replays).
__global__ void fps_init_kernel(unsigned long long* slots, unsigned* cnt, unsigned* gen) {
    const int t = threadIdx.x;
    if (t < 3)  slots[t] = 0ull;
    if (t == 3) *cnt = 0u;
    if (t == 4) *gen = 0u;
}

extern "C" void kernel_launch(void* const* d_in, const int* in_sizes, int n_in,
                              void* d_out, int out_size, void* d_ws, size_t ws_size,
                              hipStream_t stream) {
    (void)in_sizes; (void)n_in; (void)out_size; (void)ws_size;
    const float* pos = (const float*)d_in[0];
    int*         out = (int*)d_out;

    unsigned long long* slots = (unsigned long long*)d_ws;
    unsigned* cnt = (unsigned*)((char*)d_ws + 128);
    unsigned* gen = (unsigned*)((char*)d_ws + 256);

    fps_init_kernel<<<1, 32, 0, stream>>>(slots, cnt, gen);
    fps_persistent_kernel<<<NBLK, TPB, 0, stream>>>(pos, out, slots, cnt, gen);
}